// DigitCaps_87909390614796
// MI455X (gfx1250) — compile-verified
//
#include <hip/hip_runtime.h>
#include <hip/hip_bf16.h>

// DigitCaps routing for MI455X (gfx1250, wave32).
// K1: u_hat[n,b,p,o] via V_WMMA_F32_16X16X4_F32 (per-p GEMM [32x8]@[8x16]),
//     u tile staged to LDS with GLOBAL_LOAD_ASYNC_TO_LDS_B128,
//     result stored bf16 in d_ws as [n][b][p][o]  (~257 MB of d_ws).
// K2: one workgroup per (n,b); u_hat slice (196 KB bf16) async-copied into
//     LDS once, all 3 routing iterations fused, b_ij kept in registers.

#define NCAPS  40
#define PREV   6272
#define INCH   8
#define OUTCH  16
#define BATCH  32

typedef float v2f __attribute__((ext_vector_type(2)));
typedef float v8f __attribute__((ext_vector_type(8)));

// ---------------- Kernel 1: u_hat via f32 WMMA ----------------
#define PCHUNK     64
#define K1_THREADS 256
#define U_BSTRIDE  (PCHUNK * INCH + 4)   // 516 floats: bank-conflict padding

__global__ __launch_bounds__(K1_THREADS)
void uhat_wmma_kernel(const float* __restrict__ u,
                      const float* __restrict__ W,
                      __hip_bfloat16* __restrict__ uhat) {
  extern __shared__ float lds_u[];        // [BATCH][U_BSTRIDE]
  const int pc  = blockIdx.x;             // p-chunk: 0..97
  const int n   = blockIdx.y;             // 0..39
  const int p0  = pc * PCHUNK;
  const int tid = threadIdx.x;

  // Async stage of u[0:32, p0:p0+64, 0:8] into padded LDS.
  // Per-lane LDS dst address lets the DMA write the padded layout directly.
  {
    const unsigned long long ubase = (unsigned long long)(uintptr_t)u;
    const unsigned int lds0 = (unsigned int)(uintptr_t)lds_u;
    for (int e4 = tid; e4 < (BATCH * PCHUNK * INCH) / 4; e4 += K1_THREADS) {
      const int b  = e4 >> 7;             // 128 float4 per b
      const int r4 = e4 & 127;
      const unsigned int goff  = (unsigned int)(((b * PREV + p0) * INCH + r4 * 4) * 4);
      const unsigned int laddr = lds0 + (unsigned int)((b * U_BSTRIDE + r4 * 4) * 4);
      asm volatile("global_load_async_to_lds_b128 %0, %1, %2"
                   :: "v"(laddr), "v"(goff), "s"(ubase) : "memory");
    }
    asm volatile("s_wait_asynccnt 0x0" ::: "memory");
  }
  __syncthreads();

  const int wave    = tid >> 5;
  const int lane    = tid & 31;
  const int lane_lo = lane & 15;
  const int lane_hi = lane >> 4;

  for (int j = 0; j < PCHUNK / 8; ++j) {
    const int p_local = wave * 8 + j;
    const int p       = p0 + p_local;
    const float* wp   = W + ((size_t)n * PREV + p) * INCH * OUTCH;  // [8][16]

    v8f acc0 = {};  // b rows 0..15
    v8f acc1 = {};  // b rows 16..31
    #pragma unroll
    for (int ks = 0; ks < 2; ++ks) {
      const int i0 = ks * 4 + lane_hi * 2;          // K position this lane supplies
      v2f bfrag;                                    // B 4x16 row-striped fragment
      bfrag.x = wp[(i0 + 0) * OUTCH + lane_lo];
      bfrag.y = wp[(i0 + 1) * OUTCH + lane_lo];
      const float* up0 = lds_u + (lane_lo +  0) * U_BSTRIDE + p_local * INCH + i0;
      const float* up1 = lds_u + (lane_lo + 16) * U_BSTRIDE + p_local * INCH + i0;
      v2f a0; a0.x = up0[0]; a0.y = up0[1];
      v2f a1; a1.x = up1[0]; a1.y = up1[1];
      acc0 = __builtin_amdgcn_wmma_f32_16x16x4_f32(false, a0, false, bfrag,
                                                   (short)0, acc0, false, false);
      acc1 = __builtin_amdgcn_wmma_f32_16x16x4_f32(false, a1, false, bfrag,
                                                   (short)0, acc1, false, false);
    }
    // D layout: VGPR r holds (M = r + 8*lane_hi, N = lane_lo)
    __hip_bfloat16* outp = uhat + ((size_t)n * BATCH * PREV + p) * OUTCH + lane_lo;
    #pragma unroll
    for (int r = 0; r < 8; ++r) {
      const int b0 = r + lane_hi * 8;
      outp[(size_t)(b0)      * PREV * OUTCH] = __float2bfloat16(acc0[r]);
      outp[(size_t)(b0 + 16) * PREV * OUTCH] = __float2bfloat16(acc1[r]);
    }
  }
}

// ---------------- Kernel 2: fused 3-iteration routing ----------------
#define K2_THREADS 512
#define PPT 13   // ceil(6272/512) p's per thread

__global__ __launch_bounds__(K2_THREADS)
void routing_kernel(const __hip_bfloat16* __restrict__ uhat,
                    float* __restrict__ out) {
  extern __shared__ char smem_raw[];
  __hip_bfloat16* uh = (__hip_bfloat16*)smem_raw;                     // [PREV*OUTCH]
  float* red = (float*)(smem_raw + (size_t)PREV * OUTCH * 2);         // [16][18]

  const int b = blockIdx.x, n = blockIdx.y;
  const int tid  = threadIdx.x;
  const int wid  = tid >> 5;
  const int lane = tid & 31;

  // Async-stage u_hat[n][b] (196 KB bf16) into LDS; reused by all 3 iterations.
  {
    const unsigned long long sbase =
        (unsigned long long)(uintptr_t)(uhat + (size_t)(n * BATCH + b) * PREV * OUTCH);
    const unsigned int lds0 = (unsigned int)(uintptr_t)uh;
    const int n16 = PREV * OUTCH * 2 / 16;  // 12544 x 16B
    for (int e = tid; e < n16; e += K2_THREADS) {
      const unsigned int goff  = (unsigned int)e * 16u;
      const unsigned int laddr = lds0 + goff;
      asm volatile("global_load_async_to_lds_b128 %0, %1, %2"
                   :: "v"(laddr), "v"(goff), "s"(sbase) : "memory");
    }
    asm volatile("s_wait_asynccnt 0x0" ::: "memory");
  }
  __syncthreads();

  // Routing logits: each thread privately owns p = tid + k*512.
  float breg[PPT];
  #pragma unroll
  for (int k = 0; k < PPT; ++k) breg[k] = 0.0f;

  float vv[OUTCH];
  for (int it = 0; it < 3; ++it) {
    __syncthreads();  // red[] safe to overwrite (prev iter's reads done)

    // ---- softmax max over p (stable, matches jax.nn.softmax) ----
    float m = -3.402823466e38f;
    #pragma unroll
    for (int k = 0; k < PPT; ++k) {
      const int p = tid + k * K2_THREADS;
      if (p < PREV) m = fmaxf(m, breg[k]);
    }
    for (int off = 16; off; off >>= 1) m = fmaxf(m, __shfl_xor(m, off, 32));
    if (lane == 0) red[wid * 18 + 17] = m;
    __syncthreads();
    m = red[17];
    for (int w = 1; w < 16; ++w) m = fmaxf(m, red[w * 18 + 17]);

    // ---- single fused pass: Z = sum exp, s = sum exp * u_hat ----
    float z = 0.0f, acc[OUTCH];
    #pragma unroll
    for (int o = 0; o < OUTCH; ++o) acc[o] = 0.0f;
    #pragma unroll
    for (int k = 0; k < PPT; ++k) {
      const int p = tid + k * K2_THREADS;
      if (p < PREV) {
        const float wgt = __expf(breg[k] - m);
        z += wgt;
        const __hip_bfloat16* row = uh + (size_t)p * OUTCH;
        #pragma unroll
        for (int o = 0; o < OUTCH; ++o) acc[o] += wgt * __bfloat162float(row[o]);
      }
    }
    for (int off = 16; off; off >>= 1) {
      z += __shfl_xor(z, off, 32);
      #pragma unroll
      for (int o = 0; o < OUTCH; ++o) acc[o] += __shfl_xor(acc[o], off, 32);
    }
    __syncthreads();  // all reads of red m-slots complete
    if (lane == 0) {
      red[wid * 18 + 16] = z;
      #pragma unroll
      for (int o = 0; o < OUTCH; ++o) red[wid * 18 + o] = acc[o];
    }
    __syncthreads();
    float zt = 0.0f, st[OUTCH];
    #pragma unroll
    for (int o = 0; o < OUTCH; ++o) st[o] = 0.0f;
    for (int w = 0; w < 16; ++w) {
      zt += red[w * 18 + 16];
      #pragma unroll
      for (int o = 0; o < OUTCH; ++o) st[o] += red[w * 18 + o];
    }

    // ---- squash: v = s * |s| / (1 + |s|^2),  s = st / Z ----
    const float inv = 1.0f / zt;
    float nrm2 = 0.0f;
    #pragma unroll
    for (int o = 0; o < OUTCH; ++o) { const float s = st[o] * inv; nrm2 += s * s; }
    const float scale = nrm2 / ((1.0f + nrm2) * __builtin_sqrtf(nrm2));
    #pragma unroll
    for (int o = 0; o < OUTCH; ++o) vv[o] = st[o] * inv * scale;

    // ---- agreement update (skipped on last iteration, as in reference) ----
    if (it < 2) {
      #pragma unroll
      for (int k = 0; k < PPT; ++k) {
        const int p = tid + k * K2_THREADS;
        if (p < PREV) {
          const __hip_bfloat16* row = uh + (size_t)p * OUTCH;
          float a = 0.0f;
          #pragma unroll
          for (int o = 0; o < OUTCH; ++o) a += __bfloat162float(row[o]) * vv[o];
          breg[k] += a;
        }
      }
    }
  }

  if (tid < OUTCH) out[(size_t)(n * BATCH + b) * OUTCH + tid] = vv[tid];
}

// ---------------- launch ----------------
extern "C" void kernel_launch(void* const* d_in, const int* in_sizes, int n_in,
                              void* d_out, int out_size, void* d_ws, size_t ws_size,
                              hipStream_t stream) {
  const float* u = (const float*)d_in[0];   // [32, 6272, 8]
  const float* W = (const float*)d_in[1];   // [40, 6272, 8, 16]
  __hip_bfloat16* uhat = (__hip_bfloat16*)d_ws;  // [40][32][6272][16] bf16, ~257 MB

  dim3 g1(PREV / PCHUNK, NCAPS);  // (98, 40)
  size_t lds1 = (size_t)BATCH * U_BSTRIDE * sizeof(float);  // ~64.5 KB
  uhat_wmma_kernel<<<g1, K1_THREADS, lds1, stream>>>(u, W, uhat);

  dim3 g2(BATCH, NCAPS);          // (32, 40)
  size_t lds2 = (size_t)PREV * OUTCH * 2 + 16 * 18 * sizeof(float);  // ~197 KB
  routing_kernel<<<g2, K2_THREADS, lds2, stream>>>(uhat, (float*)d_out);
}